// SparseGyroidCovarianceProbe_76441827934266
// MI455X (gfx1250) — compile-verified
//
#include <hip/hip_runtime.h>
#include <hip/hip_bf16.h>
#include <math.h>

// Problem constants from the reference.
#define BATCH   32
#define SEQ     4096
#define HIDDEN  1024
#define WIN     32
#define WSTART  2032           // SEQ/2 - 16
#define NUM_EIG 8
#define SWEEPS  12             // Jacobi sweeps (quadratic convergence; ample for n=32 f32)

typedef __attribute__((ext_vector_type(2))) float v2f;
typedef __attribute__((ext_vector_type(8))) float v8f;

__global__ __launch_bounds__(128)
void sgcp_kernel(const float* __restrict__ h, float* __restrict__ out)
{
    // ---- LDS ----
    __shared__ float Cs[32][33];    // Gram -> centered -> normalized -> Jacobi-iterated (pad vs bank conflicts)
    __shared__ float Bt[32][33];    // Jacobi row-pass temp
    __shared__ float rowmean[32];
    __shared__ float invn[32];
    __shared__ int   partner[32];
    __shared__ float alC[32];
    __shared__ float beC[32];
    __shared__ float diag[32];
    __shared__ float sorted[32];
    __shared__ float scal[2];       // [0]=grand mean m, [1]=trace(C)

    const int tid  = threadIdx.x;
    const int lane = tid & 31;
    const int wave = tid >> 5;          // 0..3 (wave32)
    const int ti   = wave >> 1;         // tile row 0/1
    const int tj   = wave & 1;          // tile col 0/1
    const int half = lane >> 4;         // lane group: K-offset 0 or 2
    const int lr   = lane & 15;         // M (for A) / N (for B) within tile

    const float* Wb = h + (size_t)blockIdx.x * SEQ * HIDDEN + (size_t)WSTART * HIDDEN;

    // =========== Phase B: raw Gram G = W * W^T via f32 WMMA (16x16x4) ===========
    // A layout (ISA 7.12.2, 32-bit A 16x4): lanes 0-15 hold K=0,1; lanes 16-31 hold K=2,3.
    // B (4x16) mirrors it per-lane for the symmetric product: fixed column n = lr,
    // K = k0 + 2*half + {0,1}  ->  one contiguous float2 per lane per step, same for A.
    const float* arow = Wb + (size_t)(ti * 16 + lr) * HIDDEN + 2 * half;
    const float* brow = Wb + (size_t)(tj * 16 + lr) * HIDDEN + 2 * half;

    __builtin_prefetch(arow, 0, 0);     // global_prefetch_b8
    __builtin_prefetch(brow, 0, 0);

    v8f acc = {};
    for (int k0 = 0; k0 < HIDDEN; k0 += 4) {
        v2f a = *(const v2f*)(arow + k0);
        v2f b = *(const v2f*)(brow + k0);
        // 8 args: (neg_a, A, neg_b, B, c_mod, C, reuse_a, reuse_b)
        acc = __builtin_amdgcn_wmma_f32_16x16x4_f32(
            false, a, false, b, (short)0, acc, false, false);
    }
    // D layout: VGPR r -> M=r (lanes 0-15), M=r+8 (lanes 16-31); N = lane%16.
    for (int r = 0; r < 8; ++r)
        Cs[ti * 16 + half * 8 + r][tj * 16 + lr] = acc[r];
    __syncthreads();

    // =========== Phase C: double-center, row-normalize, trace ===========
    // Gc[w,v] = G[w,v] - r[w] - r[v] + m   ==  (W - mean)(W - mean)^T
    if (tid < 32) {
        float s = 0.f;
        for (int j = 0; j < 32; ++j) s += Cs[tid][j];
        rowmean[tid] = s * (1.0f / 32.0f);
    }
    __syncthreads();
    if (tid == 0) {
        float s = 0.f;
        for (int j = 0; j < 32; ++j) s += rowmean[j];
        scal[0] = s * (1.0f / 32.0f);
    }
    __syncthreads();
    {
        const float m = scal[0];
        for (int idx = tid; idx < 1024; idx += 128) {
            const int i = idx >> 5, j = idx & 31;
            Cs[i][j] = Cs[i][j] - rowmean[i] - rowmean[j] + m;
        }
    }
    __syncthreads();
    if (tid < 32) {
        float d = Cs[tid][tid];
        d = d > 0.f ? d : 0.f;                 // guard tiny negative from rounding
        invn[tid] = 1.0f / (sqrtf(d) + 1e-8f); // wn = wc / (||wc|| + EPS_NORM)
    }
    __syncthreads();
    for (int idx = tid; idx < 1024; idx += 128) {
        const int i = idx >> 5, j = idx & 31;
        Cs[i][j] *= invn[i] * invn[j];
    }
    __syncthreads();
    if (tid == 0) {
        float t = 0.f;
        for (int i = 0; i < 32; ++i) t += Cs[i][i];
        scal[1] = t;                            // trace(C) (rotation-invariant anyway)
    }
    __syncthreads();

    // =========== Phase D: parallel cyclic Jacobi eigensolver (Brent-Luk) ===========
    // Round r pairs: q0 = 16*r mod 31 (2*q0 == r mod 31); pair0 = (q0, 31);
    // pair k = ((q0+k) mod 31, (q0-k) mod 31), k=1..15  -> 16 disjoint pairs,
    // 31 rounds cover all 496 pairs.
    for (int sweep = 0; sweep < SWEEPS; ++sweep) {
        for (int rr = 0; rr < 31; ++rr) {
            if (tid < 16) {
                const int q0 = (16 * rr) % 31;
                int p, q;
                if (tid == 0) { p = q0; q = 31; }
                else          { p = (q0 + tid) % 31; q = (q0 - tid + 31) % 31; }
                if (p > q) { int t = p; p = q; q = t; }
                const float app = Cs[p][p], aqq = Cs[q][q], apq = Cs[p][q];
                float c = 1.f, s = 0.f;
                if (fabsf(apq) > 1e-12f) {
                    const float tau = (aqq - app) / (2.0f * apq);
                    const float t = (tau >= 0.f ? 1.f : -1.f) /
                                    (fabsf(tau) + sqrtf(1.0f + tau * tau));
                    c = 1.0f / sqrtf(1.0f + t * t);
                    s = t * c;
                }
                partner[p] = q; partner[q] = p;
                alC[p] = c;  alC[q] = c;
                beC[p] = -s; beC[q] = s;
            }
            __syncthreads();
            // Row pass: Bt = J^T * C
            for (int idx = tid; idx < 1024; idx += 128) {
                const int i = idx >> 5, j = idx & 31;
                Bt[i][j] = alC[i] * Cs[i][j] + beC[i] * Cs[partner[i]][j];
            }
            __syncthreads();
            // Col pass: C = Bt * J
            for (int idx = tid; idx < 1024; idx += 128) {
                const int i = idx >> 5, j = idx & 31;
                Cs[i][j] = alC[j] * Bt[i][j] + beC[j] * Bt[i][partner[j]];
            }
            __syncthreads();
        }
    }

    // =========== Phase E: sort eigenvalues (descending) + scalar epilogue ===========
    if (tid < 32) diag[tid] = Cs[tid][tid];
    __syncthreads();
    if (tid < 32) {
        const float v = diag[tid];
        int rk = 0;
        for (int j = 0; j < 32; ++j)
            if (diag[j] > v || (diag[j] == v && j < tid)) ++rk;
        sorted[rk] = v;
    }
    __syncthreads();
    if (tid == 0) {
        const float e0 = sorted[0];
        const float e7 = sorted[NUM_EIG - 1];
        const float e8 = sorted[NUM_EIG];
        const float spectral_gap = e7 - e8;
        const float decay_rate   = (e0 - e8) / (float)(NUM_EIG + 1);
        const float topo = fmaxf(spectral_gap / (decay_rate + 1e-8f), 0.0f);
        const float geo  = (e8 + 1e-8f) / (scal[1] + 1e-8f);
        const float g = topo + geo;
        const int b = blockIdx.x;
        out[b]      = g;     // gcve_scores
        out[32 + b] = 0.0f;  // fracture_scores
        out[64 + b] = g;     // total_pressure
    }
}

extern "C" void kernel_launch(void* const* d_in, const int* in_sizes, int n_in,
                              void* d_out, int out_size, void* d_ws, size_t ws_size,
                              hipStream_t stream) {
    (void)in_sizes; (void)n_in; (void)d_ws; (void)ws_size; (void)out_size;
    const float* h = (const float*)d_in[0];
    float* out = (float*)d_out;
    sgcp_kernel<<<dim3(BATCH), dim3(128), 0, stream>>>(h, out);
}